// PICNO_60206851555346
// MI455X (gfx1250) — compile-verified
//
#include <hip/hip_runtime.h>
#include <math.h>

// ---------------------------------------------------------------------------
// PINN loss (3rd-order forward-mode jets through a tanh MLP) for MI455X.
// Compute-bound (~0.4 TFLOP fp32 vs ~2MB HBM traffic) -> fp32 WMMA path:
//   V_WMMA_F32_16X16X4_F32, K=128 accumulated over 32 steps per 16x16 tile.
// 12 derivative channels share every weight GEMM; elementwise tanh-jet in
// between.  LDS: A/Z double buffer (padded rows) + row-major weight tile
// staged with GLOBAL_LOAD_ASYNC_TO_LDS_B128 when the toolchain exposes it.
// ---------------------------------------------------------------------------

typedef float v2f __attribute__((ext_vector_type(2)));
typedef float v4f __attribute__((ext_vector_type(4)));
typedef float v8f __attribute__((ext_vector_type(8)));
typedef int   v4i __attribute__((ext_vector_type(4)));

#define TPB   256          // 8 waves (wave32)
#define NEU   128
#define PTS   16           // points per workgroup tile (one WMMA M-tile)
#define PSTR  132          // padded row stride (floats) -> conflict-free A frags
#define KK    0.2f         // diffusion constant K
#define T0C   0.1f

#if defined(__has_builtin)
#  if __has_builtin(__builtin_amdgcn_global_load_async_to_lds_b128)
#    define USE_ASYNC_W 1
#  endif
#endif
#ifndef USE_ASYNC_W
#  define USE_ASYNC_W 0
#endif

#if USE_ASYNC_W
typedef v4i __attribute__((address_space(1))) g_v4i;   // global int4
typedef v4i __attribute__((address_space(3))) l_v4i;   // LDS int4
__device__ __forceinline__ g_v4i* as_global_v4i(const void* p) {
  return (g_v4i*)(unsigned long long)(uintptr_t)p;     // full 64-bit address
}
__device__ __forceinline__ l_v4i* as_lds_v4i(void* p) {
  // generic LDS pointers carry the LDS byte offset in the low 32 bits
  return (l_v4i*)(unsigned int)(uintptr_t)p;
}
#endif

// channels (NCH==12): 0:v 1:dx 2:dt 3:dy 4:dz 5:dxx 6:dxy 7:dxz 8:dty 9:dtz
//                     10:dxxy 11:dxxz          (v-order after [1,0,2,3] perm)
template<int NCH>
__device__ __forceinline__ void tanh_jet(const float* __restrict__ z,
                                         float* __restrict__ a) {
  float s  = tanhf(z[0]);
  float s1 = 1.0f - s * s;
  a[0] = s;
  if constexpr (NCH == 3) {            // IC pass: v, d/dy, d/dz
    a[1] = s1 * z[1];
    a[2] = s1 * z[2];
  } else if constexpr (NCH == 12) {
    float s2 = -2.0f * s * s1;
    float s3 = s1 * (6.0f * s * s - 2.0f);
    float zx = z[1], zt = z[2], zy = z[3], zz = z[4];
    a[1]  = s1 * zx;
    a[2]  = s1 * zt;
    a[3]  = s1 * zy;
    a[4]  = s1 * zz;
    a[5]  = s2 * zx * zx + s1 * z[5];
    a[6]  = s2 * zx * zy + s1 * z[6];
    a[7]  = s2 * zx * zz + s1 * z[7];
    a[8]  = s2 * zt * zy + s1 * z[8];
    a[9]  = s2 * zt * zz + s1 * z[9];
    a[10] = s3 * zx * zx * zy + s2 * (z[5] * zy + 2.0f * zx * z[6]) + s1 * z[10];
    a[11] = s3 * zx * zx * zz + s2 * (z[5] * zz + 2.0f * zx * z[7]) + s1 * z[11];
  }
}

// Forward jet pass for one 16-point tile.  A/Z: NCH*PTS*PSTR floats each,
// Wlds: NEU*NEU floats (row-major, same layout as global), out: NCH*PTS.
template<int NCH, bool SET_T0>
__device__ void mlp_forward(const float* __restrict__ Xt,
                            const float* __restrict__ W0, const float* __restrict__ b0,
                            const float* __restrict__ W1, const float* __restrict__ b1,
                            const float* __restrict__ W2, const float* __restrict__ b2,
                            const float* __restrict__ W3, const float* __restrict__ b3,
                            const float* __restrict__ W4, const float* __restrict__ b4,
                            float* A, float* Zb, float* Wlds, float* out) {
  const int tid = threadIdx.x;

  // ---- layer 0 (4 -> 128): direct affine + jet seed ----------------------
  for (int i = tid; i < PTS * NEU; i += TPB) {
    int p = i >> 7, j = i & (NEU - 1);
    const float* xr = Xt + p * 4;
    float v0 = xr[1];                       // x  (perm [1,0,2,3])
    float v1 = SET_T0 ? T0C : xr[0];        // t
    float v2 = xr[2], v3 = xr[3];           // y, z
    float w0 = W0[j], w1 = W0[NEU + j], w2 = W0[2 * NEU + j], w3 = W0[3 * NEU + j];
    float zb[12], ab[12];
    zb[0] = v0 * w0 + v1 * w1 + v2 * w2 + v3 * w3 + b0[j];
    if constexpr (NCH == 3)  { zb[1] = w2; zb[2] = w3; }
    if constexpr (NCH == 12) {
      zb[1] = w0; zb[2] = w1; zb[3] = w2; zb[4] = w3;
      #pragma unroll
      for (int c = 5; c < 12; ++c) zb[c] = 0.0f;
    }
    tanh_jet<NCH>(zb, ab);
    #pragma unroll
    for (int c = 0; c < NCH; ++c) A[c * (PTS * PSTR) + p * PSTR + j] = ab[c];
  }

  // ---- hidden layers 1..3 (128 -> 128) via WMMA --------------------------
  const float* WL[3] = {W1, W2, W3};
  const float* BL[3] = {b1, b2, b3};
  for (int l = 0; l < 3; ++l) {
    const float* W = WL[l];
    // stage W row-major into LDS (async DMA path when available)
#if USE_ASYNC_W
    for (int idx = tid * 4; idx < NEU * NEU; idx += TPB * 4) {
      __builtin_amdgcn_global_load_async_to_lds_b128(
          as_global_v4i(W + idx), as_lds_v4i(Wlds + idx), 0, 0);
    }
    asm volatile("s_wait_asynccnt 0" ::: "memory");
#else
    for (int idx = tid * 4; idx < NEU * NEU; idx += TPB * 4) {
      *(v4f*)(Wlds + idx) = *(const v4f*)(W + idx);
    }
#endif
    // warm L2 for the next layer's weights while this layer computes
    if (l < 2) __builtin_prefetch(WL[l + 1] + tid * 64, 0, 0);
    __syncthreads();   // also covers the layer-0 / previous jet writes to A

    // GEMM: Z[c](16x128) = A[c](16x128) @ W(128x128), per-channel/N-tile tasks
    {
      const int wave = tid >> 5, lane = tid & 31;
      const int mrow = lane & 15, kh = lane >> 4;     // A: lanes 16-31 hold K+2
      for (int task = wave; task < NCH * 8; task += TPB / 32) {
        const int c = task >> 3, nt = task & 7;
        const float* Ac = A + c * (PTS * PSTR) + mrow * PSTR + 2 * kh;
        const float* Bp = Wlds + 2 * kh * NEU + nt * 16 + mrow;
        v8f acc = {};
        #pragma unroll
        for (int kk = 0; kk < 32; ++kk) {
          v2f av = *(const v2f*)(Ac + 4 * kk);
          const float* bk = Bp + kk * 4 * NEU;
          v2f bv;
          bv.x = bk[0];
          bv.y = bk[NEU];
          acc = __builtin_amdgcn_wmma_f32_16x16x4_f32(
                    false, av, false, bv, (short)0, acc, false, false);
        }
        float* Zc = Zb + c * (PTS * PSTR);
        #pragma unroll
        for (int r = 0; r < 8; ++r)
          Zc[(r + 8 * kh) * PSTR + nt * 16 + mrow] = acc[r];
      }
    }
    __syncthreads();

    // elementwise: bias + tanh jet, Z -> A
    const float* bl = BL[l];
    for (int i = tid; i < PTS * NEU; i += TPB) {
      int p = i >> 7, j = i & (NEU - 1);
      float zb[12], ab[12];
      #pragma unroll
      for (int c = 0; c < NCH; ++c) zb[c] = Zb[c * (PTS * PSTR) + p * PSTR + j];
      zb[0] += bl[j];
      tanh_jet<NCH>(zb, ab);
      #pragma unroll
      for (int c = 0; c < NCH; ++c) A[c * (PTS * PSTR) + p * PSTR + j] = ab[c];
    }
    __syncthreads();
  }

  // ---- output layer (128 -> 1): per (channel, point) dot -----------------
  for (int t = tid; t < NCH * PTS; t += TPB) {
    int c = t >> 4, p = t & 15;
    const float* Ac = A + c * (PTS * PSTR) + p * PSTR;
    float acc = 0.0f;
    for (int j = 0; j < NEU; ++j) acc += Ac[j] * W4[j];
    if (c == 0) acc += b4[0];
    out[t] = acc;
  }
  __syncthreads();
}

// ws slots: 0 loss_u | 1 f1 | 2 f2 | 3 f3 | 4 u0 | 5 uy | 6 uz (sums of squares)

__global__ void __launch_bounds__(TPB, 1)
zero_ws_kernel(float* ws) {
  if (threadIdx.x < 8) ws[threadIdx.x] = 0.0f;
}

__global__ void __launch_bounds__(TPB, 1)
data_loss_kernel(const float* __restrict__ Xu, const float* __restrict__ utr,
                 const float* W0, const float* b0, const float* W1, const float* b1,
                 const float* W2, const float* b2, const float* W3, const float* b3,
                 const float* W4, const float* b4, float* ws) {
  extern __shared__ float smem[];
  constexpr int CF = 1 * PTS * PSTR;
  float* A    = smem;
  float* Zb   = A + CF;
  float* Wlds = Zb + CF;
  float* out  = Wlds + NEU * NEU;
  float* red  = out + 1 * PTS;

  const float* Xt = Xu + (size_t)blockIdx.x * PTS * 4;
  mlp_forward<1, false>(Xt, W0, b0, W1, b1, W2, b2, W3, b3, W4, b4,
                        A, Zb, Wlds, out);

  const int tid = threadIdx.x;
  if (tid < PTS) {
    float d = utr[(size_t)blockIdx.x * PTS + tid] - out[tid];
    red[tid] = d * d;
  }
  __syncthreads();
  if (tid == 0) {
    float s = 0.0f;
    for (int p = 0; p < PTS; ++p) s += red[p];
    atomicAdd(ws + 0, s);
  }
}

__global__ void __launch_bounds__(TPB, 1)
pde_ic_kernel(const float* __restrict__ Xf,
              const float* W0, const float* b0, const float* W1, const float* b1,
              const float* W2, const float* b2, const float* W3, const float* b3,
              const float* W4, const float* b4, float* ws) {
  extern __shared__ float smem[];
  const int tid = threadIdx.x;
  const float* Xt = Xf + (size_t)blockIdx.x * PTS * 4;

  // ---- pass 1: PDE residuals, 12 jet channels ----------------------------
  {
    constexpr int CF = 12 * PTS * PSTR;
    float* A    = smem;
    float* Zb   = A + CF;
    float* Wlds = Zb + CF;
    float* out  = Wlds + NEU * NEU;
    float* red  = out + 12 * PTS;

    mlp_forward<12, false>(Xt, W0, b0, W1, b1, W2, b2, W3, b3, W4, b4,
                           A, Zb, Wlds, out);

    if (tid < PTS) {
      float f1 = out[2 * PTS + tid] - KK * out[5 * PTS + tid];   // u_t - K u_xx
      float f2 = out[8 * PTS + tid] - KK * out[10 * PTS + tid];  // d/dy thereof
      float f3 = out[9 * PTS + tid] - KK * out[11 * PTS + tid];  // d/dz thereof
      red[tid]           = f1 * f1;
      red[PTS + tid]     = f2 * f2;
      red[2 * PTS + tid] = f3 * f3;
    }
    __syncthreads();
    if (tid == 0) {
      float s1 = 0.0f, s2 = 0.0f, s3 = 0.0f;
      for (int p = 0; p < PTS; ++p) {
        s1 += red[p]; s2 += red[PTS + p]; s3 += red[2 * PTS + p];
      }
      atomicAdd(ws + 1, s1);
      atomicAdd(ws + 2, s2);
      atomicAdd(ws + 3, s3);
    }
    __syncthreads();
  }

  // ---- pass 2: initial-condition terms at t = T0, 3 channels -------------
  {
    constexpr int CF = 3 * PTS * PSTR;
    float* A    = smem;
    float* Zb   = A + CF;
    float* Wlds = Zb + CF;
    float* out  = Wlds + NEU * NEU;
    float* red  = out + 3 * PTS;

    mlp_forward<3, true>(Xt, W0, b0, W1, b1, W2, b2, W3, b3, W4, b4,
                         A, Zb, Wlds, out);

    if (tid < PTS) {
      const float* xr = Xt + tid * 4;
      float x = xr[1], y = xr[2], z = xr[3];
      const float two_pi = 6.28318530717958647692f;
      float s2v = y * y + 0.2f * KK;
      float dxz = x - z;
      float E   = expf(-dxz * dxz / (2.0f * s2v));
      float sq  = sqrtf(two_pi * s2v);
      float s1v = 1.0f + 0.2f * KK;                        // SIGMA1^2 + 0.2K
      float dm  = x - 5.0f;                                // x - MU1
      float phi   = 10.0f / sq * E
                  + 10.0f / sqrtf(two_pi * s1v) * expf(-dm * dm / (2.0f * s1v));
      float phi_y = -10.0f * y / (s2v * sq) * E
                  + 10.0f * y * dxz * dxz / (s2v * s2v * sq) * E;
      float phi_z = 10.0f * dxz / (s2v * sq) * E;
      float d0 = out[tid]           - phi;
      float dy = out[PTS + tid]     - phi_y;
      float dz = out[2 * PTS + tid] - phi_z;
      red[tid]           = d0 * d0;
      red[PTS + tid]     = dy * dy;
      red[2 * PTS + tid] = dz * dz;
    }
    __syncthreads();
    if (tid == 0) {
      float s4 = 0.0f, s5 = 0.0f, s6 = 0.0f;
      for (int p = 0; p < PTS; ++p) {
        s4 += red[p]; s5 += red[PTS + p]; s6 += red[2 * PTS + p];
      }
      atomicAdd(ws + 4, s4);
      atomicAdd(ws + 5, s5);
      atomicAdd(ws + 6, s6);
    }
  }
}

__global__ void __launch_bounds__(TPB, 1)
finalize_kernel(const float* ws, float* out, float inv_nu, float inv_nf) {
  if (threadIdx.x == 0) {
    out[0] = ws[0] * inv_nu +
             (ws[1] + ws[2] + ws[3] + ws[4] + ws[5] + ws[6]) * inv_nf;
  }
}

extern "C" void kernel_launch(void* const* d_in, const int* in_sizes, int n_in,
                              void* d_out, int out_size, void* d_ws, size_t ws_size,
                              hipStream_t stream) {
  (void)n_in; (void)out_size; (void)ws_size;
  const float* Xu  = (const float*)d_in[0];
  const float* utr = (const float*)d_in[1];
  const float* Xf  = (const float*)d_in[2];
  const float* W0  = (const float*)d_in[3];
  const float* b0  = (const float*)d_in[4];
  const float* W1  = (const float*)d_in[5];
  const float* b1  = (const float*)d_in[6];
  const float* W2  = (const float*)d_in[7];
  const float* b2  = (const float*)d_in[8];
  const float* W3  = (const float*)d_in[9];
  const float* b3  = (const float*)d_in[10];
  const float* W4  = (const float*)d_in[11];
  const float* b4  = (const float*)d_in[12];
  float* ws  = (float*)d_ws;
  float* out = (float*)d_out;

  const int Nu = in_sizes[0] / 4;   // 8192
  const int Nf = in_sizes[2] / 4;   // 131072

  // dynamic LDS sizes (floats): 2 activation buffers + weight tile + out + red
  constexpr size_t smem_pde  = (size_t)(2 * 12 * PTS * PSTR + NEU * NEU
                                        + 12 * PTS + 3 * PTS) * sizeof(float);
  constexpr size_t smem_data = (size_t)(2 * 1 * PTS * PSTR + NEU * NEU
                                        + 1 * PTS + PTS) * sizeof(float);

  zero_ws_kernel<<<1, TPB, 0, stream>>>(ws);
  data_loss_kernel<<<Nu / PTS, TPB, smem_data, stream>>>(
      Xu, utr, W0, b0, W1, b1, W2, b2, W3, b3, W4, b4, ws);
  pde_ic_kernel<<<Nf / PTS, TPB, smem_pde, stream>>>(
      Xf, W0, b0, W1, b1, W2, b2, W3, b3, W4, b4, ws);
  finalize_kernel<<<1, TPB, 0, stream>>>(ws, out, 1.0f / (float)Nu,
                                         1.0f / (float)Nf);
}